// unit_GAT2_36146444764054
// MI455X (gfx1250) — compile-verified
//
#include <hip/hip_runtime.h>
#include <hip/hip_bf16.h>
#include <math.h>

#define NN 50000
#define EE 800000
#define INF_ 256
#define OUTF 32
#define H1N 8
#define SLOPE 0.2f

typedef float v2f __attribute__((ext_vector_type(2)));
typedef float v8f __attribute__((ext_vector_type(8)));

__device__ __forceinline__ float leaky(float x) { return x > 0.f ? x : SLOPE * x; }

// float atomic max via signed-int max (non-negative) / unsigned min (negative)
__device__ __forceinline__ void atomic_max_f32(float* addr, float v) {
    if (v >= 0.f) atomicMax((int*)addr, __float_as_int(v));
    else          atomicMin((unsigned int*)addr, __float_as_uint(v));
}

// ---------------- GEMM1: feat1 = X(50000x256) @ W1(256x256), FP32 WMMA ----------
// wave computes 16 rows x 64 cols; block = 4 waves -> 16 rows x 256 cols
__global__ void __launch_bounds__(128) gemm1_kernel(const float* __restrict__ X,
                                                    const float* __restrict__ W,
                                                    float* __restrict__ C) {
    const int wave = threadIdx.x >> 5;
    const int lane = threadIdx.x & 31;
    const int half = lane >> 4;      // K sub-group (0: K+0/1, 1: K+2/3)
    const int r    = lane & 15;      // row (A) / col (B) within tile
    const int row0 = blockIdx.x * 16;
    const int col0 = wave * 64;

    v8f acc[4] = {};
    const float* xrow = X + (size_t)(row0 + r) * INF_;

    for (int k = 0; k < INF_; k += 4) {
        v2f a;
        a.x = xrow[k + half * 2 + 0];
        a.y = xrow[k + half * 2 + 1];
#pragma unroll
        for (int t = 0; t < 4; ++t) {
            const int col = col0 + t * 16 + r;
            v2f b;
            b.x = W[(size_t)(k + half * 2 + 0) * 256 + col];
            b.y = W[(size_t)(k + half * 2 + 1) * 256 + col];
            acc[t] = __builtin_amdgcn_wmma_f32_16x16x4_f32(
                false, a, false, b, (short)0, acc[t], false, false);
        }
    }
#pragma unroll
    for (int t = 0; t < 4; ++t) {
#pragma unroll
        for (int j = 0; j < 8; ++j) {
            const int row = row0 + half * 8 + j;
            const int col = col0 + t * 16 + r;
            C[(size_t)row * 256 + col] = acc[t][j];
        }
    }
}

// ------------- GEMM2: feat2 = Y@W2 (256x32), out = Y@resW2 + b2 (fused) ---------
// wave: 16 rows, 32 cols of each product; block = 4 waves -> 64 rows
__global__ void __launch_bounds__(128) gemm2_kernel(const float* __restrict__ Y,
                                                    const float* __restrict__ W2,
                                                    const float* __restrict__ RW,
                                                    const float* __restrict__ b2,
                                                    float* __restrict__ F2,
                                                    float* __restrict__ OUTP,
                                                    int n) {
    const int wave = threadIdx.x >> 5;
    const int lane = threadIdx.x & 31;
    const int half = lane >> 4;
    const int r    = lane & 15;
    const int row0 = blockIdx.x * 64 + wave * 16;

    int rload = row0 + r;
    if (rload >= n) rload = n - 1;           // clamp OOB loads; stores guarded
    const float* yrow = Y + (size_t)rload * 256;

    v8f accW[2] = {};
    v8f accR[2] = {};

    for (int k = 0; k < 256; k += 4) {
        v2f a;
        a.x = yrow[k + half * 2 + 0];
        a.y = yrow[k + half * 2 + 1];
#pragma unroll
        for (int t = 0; t < 2; ++t) {
            const int col = t * 16 + r;
            v2f bw, br;
            bw.x = W2[(size_t)(k + half * 2 + 0) * 32 + col];
            bw.y = W2[(size_t)(k + half * 2 + 1) * 32 + col];
            accW[t] = __builtin_amdgcn_wmma_f32_16x16x4_f32(
                false, a, false, bw, (short)0, accW[t], false, false);
            br.x = RW[(size_t)(k + half * 2 + 0) * 32 + col];
            br.y = RW[(size_t)(k + half * 2 + 1) * 32 + col];
            accR[t] = __builtin_amdgcn_wmma_f32_16x16x4_f32(
                false, a, false, br, (short)0, accR[t], false, false);
        }
    }
#pragma unroll
    for (int t = 0; t < 2; ++t) {
#pragma unroll
        for (int j = 0; j < 8; ++j) {
            const int row = row0 + half * 8 + j;
            const int col = t * 16 + r;
            if (row < n) {
                F2[(size_t)row * 32 + col]   = accW[t][j];
                OUTP[(size_t)row * 32 + col] = accR[t][j] + b2[col];
            }
        }
    }
}

// -------- per-node attention logits, layer 1: wave32 == one head (32 elems) -----
__global__ void __launch_bounds__(256) elr1_kernel(const float* __restrict__ feat,
                                                   const float* __restrict__ al,
                                                   const float* __restrict__ ar,
                                                   float* __restrict__ el,
                                                   float* __restrict__ er) {
    const int n = blockIdx.x;
    const int d = threadIdx.x;            // 0..255, wave w == head w
    const float v = feat[(size_t)n * 256 + d];
    float pl = v * al[d];
    float pr = v * ar[d];
#pragma unroll
    for (int off = 16; off > 0; off >>= 1) {
        pl += __shfl_xor(pl, off, 32);
        pr += __shfl_xor(pr, off, 32);
    }
    if ((d & 31) == 0) {
        const int h = d >> 5;
        el[n * H1N + h] = pl;
        er[n * H1N + h] = pr;
    }
}

// -------- per-node logits, layer 2: wave32 == one node (32 feats) ---------------
__global__ void __launch_bounds__(256) elr2_kernel(const float* __restrict__ feat,
                                                   const float* __restrict__ al,
                                                   const float* __restrict__ ar,
                                                   float* __restrict__ el,
                                                   float* __restrict__ er,
                                                   int n) {
    const int node = blockIdx.x * 8 + (threadIdx.x >> 5);
    const int lane = threadIdx.x & 31;
    if (node >= n) return;
    const float v = feat[(size_t)node * 32 + lane];
    float pl = v * al[lane];
    float pr = v * ar[lane];
#pragma unroll
    for (int off = 16; off > 0; off >>= 1) {
        pl += __shfl_xor(pl, off, 32);
        pr += __shfl_xor(pr, off, 32);
    }
    if (lane == 0) { el[node] = pl; er[node] = pr; }
}

// ----------------------------- init helpers -------------------------------------
__global__ void __launch_bounds__(256) init_rst1_kernel(const float* __restrict__ x,
                                                        const float* __restrict__ b1,
                                                        float* __restrict__ rst) {
    const size_t i = (size_t)blockIdx.x * 256 + threadIdx.x;   // grid == NN blocks
    rst[i] = x[i] + b1[threadIdx.x];
}

__global__ void __launch_bounds__(256) init_softmax_kernel(float* __restrict__ emax,
                                                           float* __restrict__ esum,
                                                           int n) {
    const int i = blockIdx.x * 256 + threadIdx.x;
    if (i < n) { emax[i] = -INFINITY; esum[i] = 0.f; }
}

__global__ void __launch_bounds__(256) relu_kernel(float* __restrict__ y) {
    const size_t i = (size_t)blockIdx.x * 256 + threadIdx.x;   // grid == NN blocks
    y[i] = fmaxf(y[i], 0.f);
}

// ------------------------- edge softmax passes, layer 1 -------------------------
__global__ void __launch_bounds__(256) edge1_max_kernel(const int* __restrict__ src,
                                                        const int* __restrict__ dst,
                                                        const float* __restrict__ el,
                                                        const float* __restrict__ er,
                                                        float* __restrict__ emax) {
    const int tid = blockIdx.x * 256 + threadIdx.x;
    if (tid >= EE * H1N) return;
    const int e = tid >> 3, h = tid & 7;
    const int s = src[e], d = dst[e];
    const float sc = leaky(el[s * H1N + h] + er[d * H1N + h]);
    atomic_max_f32(&emax[d * H1N + h], sc);
}

__global__ void __launch_bounds__(256) edge1_sum_kernel(const int* __restrict__ src,
                                                        const int* __restrict__ dst,
                                                        const float* __restrict__ el,
                                                        const float* __restrict__ er,
                                                        const float* __restrict__ emax,
                                                        float* __restrict__ esum) {
    const int tid = blockIdx.x * 256 + threadIdx.x;
    if (tid >= EE * H1N) return;
    const int e = tid >> 3, h = tid & 7;
    const int s = src[e], d = dst[e];
    const float sc = leaky(el[s * H1N + h] + er[d * H1N + h]);
    atomicAdd(&esum[d * H1N + h], __expf(sc - emax[d * H1N + h]));
}

// wave32 per edge; lane moves 8 floats (head = lane/4)
__global__ void __launch_bounds__(256) edge1_agg_kernel(const int* __restrict__ src,
                                                        const int* __restrict__ dst,
                                                        const float* __restrict__ el,
                                                        const float* __restrict__ er,
                                                        const float* __restrict__ emax,
                                                        const float* __restrict__ esum,
                                                        const float* __restrict__ feat,
                                                        float* __restrict__ rst) {
    const int gid  = blockIdx.x * 256 + threadIdx.x;
    const int e    = gid >> 5;
    if (e >= EE) return;
    const int lane = gid & 31;
    const int s = src[e], d = dst[e];
    const int h = lane >> 2;
    const float sc = leaky(el[s * H1N + h] + er[d * H1N + h]);
    const float a  = __expf(sc - emax[d * H1N + h]) / esum[d * H1N + h];

    const float4* fs = (const float4*)(feat + (size_t)s * 256 + lane * 8);
    const float4 f0 = fs[0], f1 = fs[1];
    float* rd = rst + (size_t)d * 256 + lane * 8;
    atomicAdd(rd + 0, f0.x * a); atomicAdd(rd + 1, f0.y * a);
    atomicAdd(rd + 2, f0.z * a); atomicAdd(rd + 3, f0.w * a);
    atomicAdd(rd + 4, f1.x * a); atomicAdd(rd + 5, f1.y * a);
    atomicAdd(rd + 6, f1.z * a); atomicAdd(rd + 7, f1.w * a);
}

// ------------------------- edge softmax passes, layer 2 -------------------------
__global__ void __launch_bounds__(256) edge2_max_kernel(const int* __restrict__ src,
                                                        const int* __restrict__ dst,
                                                        const float* __restrict__ el,
                                                        const float* __restrict__ er,
                                                        float* __restrict__ emax) {
    const int e = blockIdx.x * 256 + threadIdx.x;
    if (e >= EE) return;
    const int s = src[e], d = dst[e];
    atomic_max_f32(&emax[d], leaky(el[s] + er[d]));
}

__global__ void __launch_bounds__(256) edge2_sum_kernel(const int* __restrict__ src,
                                                        const int* __restrict__ dst,
                                                        const float* __restrict__ el,
                                                        const float* __restrict__ er,
                                                        const float* __restrict__ emax,
                                                        float* __restrict__ esum) {
    const int e = blockIdx.x * 256 + threadIdx.x;
    if (e >= EE) return;
    const int s = src[e], d = dst[e];
    const float sc = leaky(el[s] + er[d]);
    atomicAdd(&esum[d], __expf(sc - emax[d]));
}

__global__ void __launch_bounds__(256) edge2_agg_kernel(const int* __restrict__ src,
                                                        const int* __restrict__ dst,
                                                        const float* __restrict__ el,
                                                        const float* __restrict__ er,
                                                        const float* __restrict__ emax,
                                                        const float* __restrict__ esum,
                                                        const float* __restrict__ feat,
                                                        float* __restrict__ outp) {
    const int gid  = blockIdx.x * 256 + threadIdx.x;
    const int e    = gid >> 5;
    if (e >= EE) return;
    const int lane = gid & 31;
    const int s = src[e], d = dst[e];
    const float sc = leaky(el[s] + er[d]);
    const float a  = __expf(sc - emax[d]) / esum[d];
    atomicAdd(outp + (size_t)d * 32 + lane, feat[(size_t)s * 32 + lane] * a);
}

// --------------------------------------------------------------------------------
extern "C" void kernel_launch(void* const* d_in, const int* in_sizes, int n_in,
                              void* d_out, int out_size, void* d_ws, size_t ws_size,
                              hipStream_t stream) {
    const float* x     = (const float*)d_in[0];
    const int*   src   = (const int*)  d_in[1];
    const int*   dst   = (const int*)  d_in[2];
    const float* W1    = (const float*)d_in[3];
    const float* al1   = (const float*)d_in[4];   // (8,32) flat == 256
    const float* ar1   = (const float*)d_in[5];
    const float* b1    = (const float*)d_in[6];
    const float* W2    = (const float*)d_in[7];
    const float* al2   = (const float*)d_in[8];   // (1,32) flat == 32
    const float* ar2   = (const float*)d_in[9];
    const float* resW2 = (const float*)d_in[10];
    const float* b2    = (const float*)d_in[11];
    float* outp = (float*)d_out;

    float* ws    = (float*)d_ws;
    float* feat1 = ws;                         // NN*256
    float* rst1  = feat1 + (size_t)NN * 256;   // NN*256 (becomes y1 after ReLU)
    float* el1   = rst1  + (size_t)NN * 256;   // NN*8
    float* er1   = el1   + (size_t)NN * 8;
    float* emax1 = er1   + (size_t)NN * 8;
    float* esum1 = emax1 + (size_t)NN * 8;
    float* feat2 = esum1 + (size_t)NN * 8;     // NN*32
    float* el2   = feat2 + (size_t)NN * 32;    // NN
    float* er2   = el2   + NN;
    float* emax2 = er2   + NN;
    float* esum2 = emax2 + NN;

    // ---- layer 1 ----
    gemm1_kernel<<<NN / 16, 128, 0, stream>>>(x, W1, feat1);
    init_rst1_kernel<<<NN, 256, 0, stream>>>(x, b1, rst1);          // identity residual + bias
    init_softmax_kernel<<<(NN * H1N + 255) / 256, 256, 0, stream>>>(emax1, esum1, NN * H1N);
    elr1_kernel<<<NN, 256, 0, stream>>>(feat1, al1, ar1, el1, er1);
    edge1_max_kernel<<<(EE * H1N + 255) / 256, 256, 0, stream>>>(src, dst, el1, er1, emax1);
    edge1_sum_kernel<<<(EE * H1N + 255) / 256, 256, 0, stream>>>(src, dst, el1, er1, emax1, esum1);
    edge1_agg_kernel<<<(EE * 32) / 256, 256, 0, stream>>>(src, dst, el1, er1, emax1, esum1,
                                                          feat1, rst1);
    relu_kernel<<<NN, 256, 0, stream>>>(rst1);                      // rst1 -> y1

    // ---- layer 2 ----
    init_softmax_kernel<<<(NN + 255) / 256, 256, 0, stream>>>(emax2, esum2, NN);
    gemm2_kernel<<<(NN + 63) / 64, 128, 0, stream>>>(rst1, W2, resW2, b2, feat2, outp, NN);
    elr2_kernel<<<(NN + 7) / 8, 256, 0, stream>>>(feat2, al2, ar2, el2, er2, NN);
    edge2_max_kernel<<<(EE + 255) / 256, 256, 0, stream>>>(src, dst, el2, er2, emax2);
    edge2_sum_kernel<<<(EE + 255) / 256, 256, 0, stream>>>(src, dst, el2, er2, emax2, esum2);
    edge2_agg_kernel<<<(EE * 32) / 256, 256, 0, stream>>>(src, dst, el2, er2, emax2, esum2,
                                                          feat2, outp);
}